// MQCNN_layer_44229573214628
// MI455X (gfx1250) — compile-verified
//
#include <hip/hip_runtime.h>
#include <hip/hip_bf16.h>

typedef __attribute__((ext_vector_type(2))) float v2f;
typedef __attribute__((ext_vector_type(8))) float v8f;

#define WSTR 132   // padded LDS stride for W (32 x 128 patch-columns)
#define GSTR 33    // padded LDS stride for G (32 x 32)

// ---------------------------------------------------------------------------
// Kernel A: build the 32x32 real effective measurement matrix G from U3_w.
//
// The conv circuit acts on wires {0,2,4,5,6}; subspace index s (5 bits):
//   wire0 -> s bit4, wire2 -> s bit3, wire4 -> s bit2, wire5 -> s bit1,
//   wire6 -> s bit0 (target of all U3 gates).
// U = (24 controlled-U3) * H(wire5).  K = U^H * Z4 * U (32x32 Hermitian).
// Encoded states live on wires5,6 = 0, i.e. s = q<<2, q in [0,8).
// Kr[q][q'] = Re K[q<<2][q'<<2] (8x8 symmetric).  Expanded over bystander
// wires 1,3:  G[j][j'] = (c1==c1' && c3==c3') ? Kr[q][q'] : 0,  j 5 bits
// over wires (0,1,2,3,4) -> bits (4,3,2,1,0).
// ---------------------------------------------------------------------------
__global__ __launch_bounds__(256) void mqcnn_build_G(const float* __restrict__ w3,
                                                     float* __restrict__ G)
{
    __shared__ float sRe[32 * 32];
    __shared__ float sIm[32 * 32];
    __shared__ float sK[64];

    const int tid = threadIdx.x;

    if (tid < 32) {
        // simulate column t = tid of the conv unitary in registers
        float re[32], im[32];
        #pragma unroll
        for (int s = 0; s < 32; ++s) { re[s] = (s == tid) ? 1.0f : 0.0f; im[s] = 0.0f; }

        // H on wire5 (s bit1)
        const float r2 = 0.70710678118654752f;
        #pragma unroll
        for (int s = 0; s < 32; ++s) {
            if ((s & 2) == 0) {
                const int s1 = s | 2;
                const float ar = re[s], ai = im[s], br = re[s1], bi = im[s1];
                re[s]  = (ar + br) * r2;  im[s]  = (ai + bi) * r2;
                re[s1] = (ar - br) * r2;  im[s1] = (ai - bi) * r2;
            }
        }

        // control masks for the 8-gate block (wire4->bit2, wire0->bit4,
        // wire2->bit3, wire5->bit1), target = bit0
        const int cmask[8] = {0x04, 0x14, 0x0C, 0x1C, 0x06, 0x16, 0x0E, 0x1E};

        #pragma unroll
        for (int g = 0; g < 24; ++g) {
            const float th = w3[g * 3 + 0] * 0.5f;
            const float ph = w3[g * 3 + 1];
            const float la = w3[g * 3 + 2];
            float st, ct;  __sincosf(th, &st, &ct);
            float sl, cl;  __sincosf(la, &sl, &cl);
            float sp, cp;  __sincosf(ph, &sp, &cp);
            float spl, cpl; __sincosf(ph + la, &spl, &cpl);
            // U3 = [[ct, -e^{i la} st], [e^{i ph} st, e^{i(ph+la)} ct]]
            const float u00r = ct,        u00i = 0.0f;
            const float u01r = -cl * st,  u01i = -sl * st;
            const float u10r =  cp * st,  u10i =  sp * st;
            const float u11r = cpl * ct,  u11i = spl * ct;
            const int mask = cmask[g & 7];   // compile-time after unroll

            #pragma unroll
            for (int s0 = 0; s0 < 32; ++s0) {
                if (((s0 & 1) == 0) && ((s0 & mask) == mask)) {
                    const int s1 = s0 | 1;
                    const float ar = re[s0], ai = im[s0], br = re[s1], bi = im[s1];
                    re[s0] = u00r * ar - u00i * ai + u01r * br - u01i * bi;
                    im[s0] = u00r * ai + u00i * ar + u01r * bi + u01i * br;
                    re[s1] = u10r * ar - u10i * ai + u11r * br - u11i * bi;
                    im[s1] = u10r * ai + u10i * ar + u11r * bi + u11i * br;
                }
            }
        }

        #pragma unroll
        for (int s = 0; s < 32; ++s) { sRe[s * 32 + tid] = re[s]; sIm[s * 32 + tid] = im[s]; }
    }
    __syncthreads();

    if (tid < 64) {
        const int q  = tid >> 3;
        const int qp = tid & 7;
        float acc = 0.0f;
        for (int s = 0; s < 32; ++s) {
            const float zs = ((s >> 2) & 1) ? -1.0f : 1.0f;   // Z on wire4 (s bit2)
            acc += zs * (sRe[s * 32 + (q << 2)] * sRe[s * 32 + (qp << 2)] +
                         sIm[s * 32 + (q << 2)] * sIm[s * 32 + (qp << 2)]);
        }
        sK[tid] = acc;
    }
    __syncthreads();

    for (int e = tid; e < 1024; e += 256) {
        const int j = e >> 5, jp = e & 31;
        const int c1 = (j >> 3) & 1, c3 = (j >> 1) & 1;
        const int c1p = (jp >> 3) & 1, c3p = (jp >> 1) & 1;
        float val = 0.0f;
        if (c1 == c1p && c3 == c3p) {
            const int q  = (((j  >> 4) & 1) << 2) | (((j  >> 2) & 1) << 1) | (j  & 1);
            const int qp = (((jp >> 4) & 1) << 2) | (((jp >> 2) & 1) << 1) | (jp & 1);
            val = sK[q * 8 + qp];
        }
        G[e] = val;
    }
}

// ---------------------------------------------------------------------------
// Kernel B: per patch, build the real 32-entry encoded state w, then compute
// z_p = w^T G w via batched WMMA (Y = G*W tiles), dot, and reduce.
// 256 threads = 8 waves; block handles 128 patches; wave handles 16 patches.
// ---------------------------------------------------------------------------
__global__ __launch_bounds__(256) void mqcnn_main(const float* __restrict__ x,
                                                  const float* __restrict__ G,
                                                  float* __restrict__ out)
{
    __shared__ float Wl[32 * WSTR];
    __shared__ float Gl[32 * GSTR];

    const int tid = threadIdx.x;

    // stage G into LDS
    for (int e = tid; e < 1024; e += 256)
        Gl[(e >> 5) * GSTR + (e & 31)] = G[e];

    // ---- phase 1: encoding (2 threads per patch, 8 configs each) ----
    const int pBlock = blockIdx.x * 128;
    const int pi   = tid >> 1;      // patch within block: 0..127
    const int half = tid & 1;       // c0 bit
    const int p  = pBlock + pi;
    const int b  = p >> 12;
    const int jk = p & 4095;
    const int jj = jk >> 6;
    const int kk = jk & 63;

    const float* xb = x + (size_t)b * 3 * 16384 + (2 * jj) * 128 + 2 * kk;
    float pv[12];
    #pragma unroll
    for (int ch = 0; ch < 3; ++ch) {
        const float2 r0 = *(const float2*)(xb + ch * 16384);
        const float2 r1 = *(const float2*)(xb + ch * 16384 + 128);
        pv[ch * 4 + 0] = r0.x; pv[ch * 4 + 1] = r0.y;
        pv[ch * 4 + 2] = r1.x; pv[ch * 4 + 3] = r1.y;
    }
    const float A   = pv[0] + pv[4] + pv[8];
    const float B01 = pv[3];
    const float B02 = pv[1] + pv[6];
    const float B03 = pv[5] + pv[10];
    const float B13 = pv[2] + pv[11];
    const float B23 = pv[7] + pv[9];

    #pragma unroll
    for (int cc = 0; cc < 8; ++cc) {           // c0 = half, c1..c3 = cc bits
        const int c1 = (cc >> 2) & 1, c2 = (cc >> 1) & 1, c3 = cc & 1;
        float th = A;
        if (c1 && c3) th += B13;
        if (c2 && c3) th += B23;
        if (half) {
            if (c1) th += B01;
            if (c2) th += B02;
            if (c3) th += B03;
        }
        float sn, cn;
        __sincosf(0.5f * th, &sn, &cn);
        const int cidx = (half << 3) | cc;     // j = cidx*2 + b4
        Wl[(2 * cidx + 0) * WSTR + pi] = 0.25f * cn;
        Wl[(2 * cidx + 1) * WSTR + pi] = 0.25f * sn;
    }
    __syncthreads();

    // ---- phase 2: Y = G * W via v_wmma_f32_16x16x4_f32 ----
    const int lane = tid & 31;
    const int wv   = tid >> 5;
    const int col  = lane & 15;
    const int hi   = lane >> 4;      // 0: K=0,1 / M rows 0-7 ; 1: K=2,3 / M rows 8-15
    const int p16  = wv * 16;

    v8f acc0 = {}; v8f acc1 = {};
    #pragma unroll
    for (int t = 0; t < 8; ++t) {
        const int kc = t * 4 + hi * 2;
        v2f bf, a0, a1;
        bf.x = Wl[(kc + 0) * WSTR + p16 + col];
        bf.y = Wl[(kc + 1) * WSTR + p16 + col];
        a0.x = Gl[col * GSTR + kc];            // G rows 0..15
        a0.y = Gl[col * GSTR + kc + 1];
        a1.x = Gl[(16 + col) * GSTR + kc];     // G rows 16..31
        a1.y = Gl[(16 + col) * GSTR + kc + 1];
        acc0 = __builtin_amdgcn_wmma_f32_16x16x4_f32(false, a0, false, bf,
                                                     (short)0, acc0, false, false);
        acc1 = __builtin_amdgcn_wmma_f32_16x16x4_f32(false, a1, false, bf,
                                                     (short)0, acc1, false, false);
    }

    // ---- phase 3: z_p = sum_j W[j,p] * Y[j,p], reduce across lane pairs ----
    float zp = 0.0f;
    #pragma unroll
    for (int v = 0; v < 8; ++v) {
        zp += acc0[v] * Wl[(v + 8 * hi)      * WSTR + p16 + col];
        zp += acc1[v] * Wl[(16 + v + 8 * hi) * WSTR + p16 + col];
    }
    zp += __shfl_xor(zp, 16, 32);
    if (hi == 0) out[pBlock + p16 + col] = zp;
}

extern "C" void kernel_launch(void* const* d_in, const int* in_sizes, int n_in,
                              void* d_out, int out_size, void* d_ws, size_t ws_size,
                              hipStream_t stream) {
    const float* x  = (const float*)d_in[0];   // [32,3,128,128] f32
    const float* w3 = (const float*)d_in[1];   // [24,3] f32
    float* G   = (float*)d_ws;                 // 32x32 f32 scratch
    float* out = (float*)d_out;                // [32,1,64,64] f32 = 131072

    mqcnn_build_G<<<1, 256, 0, stream>>>(w3, G);
    mqcnn_main<<<1024, 256, 0, stream>>>(x, G, out);
}